// EfficientMemorySiLU_54451595379275
// MI455X (gfx1250) — compile-verified
//
#include <hip/hip_runtime.h>

// ---------------------------------------------------------------------------
// EfficientMemorySiLU for MI455X (gfx1250), compile-only tuning.
//
// d_out layout (float32): [ silu(x) : n ][ outliner : 1 ][ scale : 1 ]
//
// Plan:
//  * 3-pass radix-select (10/11/10 bits of abs-bit-pattern) over x[0] for the
//    exact k-th order statistic (k = int(n0*0.99), 1-based). abs-f32 bit
//    patterns are monotone, so integer bucket counting gives the exact float.
//  * scale = outliner / 127 (bit-exact: outliner is itself an element of
//    |x[0]|, so max(|x_sub0|) == outliner).
//  * SiLU kernel: HBM-bound streaming (~512 MiB / 23.3 TB/s ~= 22 us floor);
//    uses gfx1250 async global->LDS loads (ASYNCcnt) with a 2-deep per-lane
//    double buffer + split wait counters, nontemporal b128 stores.
//    iters is passed from the host so no runtime integer division remains.
// ---------------------------------------------------------------------------

typedef float        f32x4 __attribute__((ext_vector_type(4)));
typedef unsigned int u32x4 __attribute__((ext_vector_type(4)));
typedef unsigned int uint32;
// Exact parameter type of __builtin_amdgcn_global_load_async_to_lds_b128:
// 'int __attribute__((vector_size(16)))' pointers in AS1 (global) / AS3 (LDS).
typedef int v4i __attribute__((vector_size(16)));

#define AS1 __attribute__((address_space(1)))
#define AS3 __attribute__((address_space(3)))

__device__ __forceinline__ float silu1(float v) {
  return v / (1.0f + __expf(-v));
}

// ---------------------------------------------------------------------------
// SiLU streaming kernel (grid-stride over float4), async-LDS double buffered.
// `iters` = nvec / (gridDim.x*blockDim.x), computed on the host (uniform).
// ---------------------------------------------------------------------------
#define SILU_BLOCK 256

__global__ __launch_bounds__(SILU_BLOCK) void silu_kernel(
    const f32x4* __restrict__ x, f32x4* __restrict__ out, int nvec, int iters) {
  const int tid    = (int)threadIdx.x;
  const int gid    = (int)blockIdx.x * SILU_BLOCK + tid;
  const int stride = (int)gridDim.x * SILU_BLOCK;

#if defined(__gfx1250__) && __has_builtin(__builtin_amdgcn_global_load_async_to_lds_b128)
  // Per-lane staging: each lane consumes exactly the 16B it fetched, so no
  // cross-wave barrier is needed; visibility is via ASYNCcnt only.
  __shared__ f32x4 stage[2][SILU_BLOCK];

  if (iters > 0) {
    __builtin_amdgcn_global_load_async_to_lds_b128(
        (AS1 v4i*)(x + gid), (AS3 v4i*)&stage[0][tid], 0, 0);
  }
  for (int i = 0; i < iters; ++i) {
    if (i + 1 < iters) {
      // Ensure the ds_load that previously read the buffer we are about to
      // overwrite has fully returned before the async engine can write it.
      asm volatile("s_wait_dscnt 0x0" ::: "memory");
      __builtin_amdgcn_global_load_async_to_lds_b128(
          (AS1 v4i*)(x + gid + (i + 1) * stride),
          (AS3 v4i*)&stage[(i + 1) & 1][tid], 0, 0);
      asm volatile("s_wait_asynccnt 0x1" ::: "memory");  // buffer i ready
    } else {
      asm volatile("s_wait_asynccnt 0x0" ::: "memory");  // last buffer ready
    }
    f32x4 v = stage[i & 1][tid];
    f32x4 r;
    r.x = silu1(v.x); r.y = silu1(v.y); r.z = silu1(v.z); r.w = silu1(v.w);
    __builtin_nontemporal_store(r, out + gid + i * stride);
  }
  // Remainder (empty for the reference shape: 16Mi float4 % (8192*256) == 0)
  for (int idx = iters * stride + gid; idx < nvec; idx += stride) {
    f32x4 v = x[idx];
    f32x4 r;
    r.x = silu1(v.x); r.y = silu1(v.y); r.z = silu1(v.z); r.w = silu1(v.w);
    __builtin_nontemporal_store(r, out + idx);
  }
#else
  for (int idx = gid; idx < nvec; idx += stride) {
    f32x4 v = x[idx];
    f32x4 r;
    r.x = silu1(v.x); r.y = silu1(v.y); r.z = silu1(v.z); r.w = silu1(v.w);
    __builtin_nontemporal_store(r, out + idx);
  }
#endif
}

// ---------------------------------------------------------------------------
// Histogram pass over abs-bit-patterns of x[0] (radix-select level).
//  level0: prefIdx=-1,              bin = key >> 21          (1024 bins)
//  level1: prefIdx=0, prefShift=21, bin = (key>>10)&0x7FF    (2048 bins)
//  level2: prefIdx=2, prefShift=10, bin = key & 0x3FF        (1024 bins)
// ---------------------------------------------------------------------------
#define HIST_BLOCK 256

__global__ __launch_bounds__(HIST_BLOCK) void hist_kernel(
    const u32x4* __restrict__ xb, int nvec, uint32* __restrict__ bins,
    int nbins, int binShift, uint32 binMask, int prefShift,
    const uint32* __restrict__ ctrl, int prefIdx) {
  __shared__ uint32 h[2048];
  const int tid = (int)threadIdx.x;
  for (int i = tid; i < nbins; i += HIST_BLOCK) h[i] = 0u;
  __syncthreads();

  const bool matchAll = (prefIdx < 0);
  uint32 pref = 0u;
  if (!matchAll) pref = ctrl[prefIdx];  // uniform scalar load

  const int gid    = (int)blockIdx.x * HIST_BLOCK + tid;
  const int stride = (int)gridDim.x * HIST_BLOCK;
  for (int i = gid; i < nvec; i += stride) {
    u32x4 k4 = xb[i];
#pragma unroll
    for (int c = 0; c < 4; ++c) {
      uint32 key = k4[c] & 0x7FFFFFFFu;  // |x| bit pattern (monotone)
      if (matchAll || ((key >> prefShift) == pref)) {
        atomicAdd(&h[(key >> binShift) & binMask], 1u);  // ds_add_u32
      }
    }
  }
  __syncthreads();
  for (int i = tid; i < nbins; i += HIST_BLOCK) {
    uint32 cnt = h[i];
    if (cnt) atomicAdd(&bins[i], cnt);
  }
}

// ---------------------------------------------------------------------------
// Single-block scan: find bin b with prefix(b-1) < r <= prefix(b).
// mode 0: r=K       -> ctrl[0]=b,                ctrl[1]=r'
// mode 1: r=ctrl[1] -> ctrl[2]=(ctrl[0]<<11)|b,  ctrl[3]=r'
// mode 2: r=ctrl[3] -> key=(ctrl[2]<<10)|b; write outliner & scale.
// ---------------------------------------------------------------------------
template <int NBINS>
__global__ __launch_bounds__(256) void select_kernel(
    const uint32* __restrict__ bins, int mode, int K,
    uint32* __restrict__ ctrl, float* __restrict__ outTail) {
  constexpr int CHUNK = NBINS / 256;
  const int t = (int)threadIdx.x;

  uint32 loc[CHUNK];
  uint32 s = 0u;
#pragma unroll
  for (int j = 0; j < CHUNK; ++j) {
    loc[j] = bins[t * CHUNK + j];
    s += loc[j];
  }

  __shared__ uint32 ps[256];
  ps[t] = s;
  __syncthreads();
  for (int off = 1; off < 256; off <<= 1) {  // Hillis-Steele inclusive scan
    uint32 v = (t >= off) ? ps[t - off] : 0u;
    __syncthreads();
    ps[t] += v;
    __syncthreads();
  }
  const uint32 excl = ps[t] - s;  // count in all bins before this chunk

  uint32 r;
  if (mode == 0)      r = (uint32)K;
  else if (mode == 1) r = ctrl[1];
  else                r = ctrl[3];

  uint32 c = excl;
#pragma unroll
  for (int j = 0; j < CHUNK; ++j) {
    uint32 nb = c + loc[j];
    if (c < r && r <= nb) {  // exactly one (bin, thread) satisfies this
      uint32 b    = (uint32)(t * CHUNK + j);
      uint32 rNew = r - c;
      if (mode == 0) {
        ctrl[0] = b;
        ctrl[1] = rNew;
      } else if (mode == 1) {
        ctrl[2] = (ctrl[0] << 11) | b;
        ctrl[3] = rNew;
      } else {
        uint32 key = (ctrl[2] << 10) | b;  // exact 31-bit abs pattern
        float  o   = __uint_as_float(key);
        outTail[0] = o;            // outliner
        outTail[1] = o / 127.0f;   // scale == max(|x_sub0|)/127 exactly
      }
    }
    c = nb;
  }
}

// ---------------------------------------------------------------------------
extern "C" void kernel_launch(void* const* d_in, const int* in_sizes, int n_in,
                              void* d_out, int out_size, void* d_ws, size_t ws_size,
                              hipStream_t stream) {
  (void)n_in; (void)out_size; (void)ws_size;

  const float* x   = (const float*)d_in[0];
  float*       out = (float*)d_out;

  const int n  = in_sizes[0];  // 8*2048*4096 = 67108864
  const int n0 = n / 8;        // elements in x[0]
  // k = int(n0 * (1.0 - 0.01)) — matches Python int() truncation semantics.
  const int K  = (int)((double)n0 * (1.0 - 0.01));  // 1-based rank

  // Workspace layout (uint32): bins0[1024] bins1[2048] bins2[1024] ctrl[8]
  uint32* ws    = (uint32*)d_ws;
  uint32* bins0 = ws;
  uint32* bins1 = ws + 1024;
  uint32* bins2 = ws + 1024 + 2048;
  uint32* ctrl  = ws + 1024 + 2048 + 1024;
  (void)hipMemsetAsync(d_ws, 0, (1024 + 2048 + 1024 + 8) * sizeof(uint32), stream);

  const int    nvec0 = n0 / 4;
  const u32x4* x0v   = (const u32x4*)x;
  float*       tailp = out + n;

  // Selection first so x[0] (32 MiB) stays L2-resident across the 3 passes;
  // the 512 MiB silu stream runs last.
  hist_kernel<<<1024, HIST_BLOCK, 0, stream>>>(x0v, nvec0, bins0, 1024, 21, 0x3FFu, 0, ctrl, -1);
  select_kernel<1024><<<1, 256, 0, stream>>>(bins0, 0, K, ctrl, tailp);
  hist_kernel<<<1024, HIST_BLOCK, 0, stream>>>(x0v, nvec0, bins1, 2048, 10, 0x7FFu, 21, ctrl, 0);
  select_kernel<2048><<<1, 256, 0, stream>>>(bins1, 1, K, ctrl, tailp);
  hist_kernel<<<1024, HIST_BLOCK, 0, stream>>>(x0v, nvec0, bins2, 1024, 0, 0x3FFu, 10, ctrl, 2);
  select_kernel<1024><<<1, 256, 0, stream>>>(bins2, 2, K, ctrl, tailp);

  // 8192 blocks * 256 threads; for n = 64Mi -> nvec = 16Mi -> 8 uniform iters.
  const int silu_blocks = 8192;
  const int nvec        = n / 4;
  const int stride      = silu_blocks * SILU_BLOCK;
  const int iters       = nvec / stride;
  silu_kernel<<<silu_blocks, SILU_BLOCK, 0, stream>>>((const f32x4*)x, (f32x4*)out, nvec, iters);
}